// TSPVectorPopGraph_85057532330019
// MI455X (gfx1250) — compile-verified
//
#include <hip/hip_runtime.h>
#include <hip/hip_bf16.h>
#include <math.h>

// ---------------------------------------------------------------------------
// MI455X / gfx1250 implementation of the TSP-vector-pop-graph pipeline.
// wave32 everywhere. All heavy matmuls (kNN Gram matrix + GCN layers) run on
// V_WMMA_F32_16X16X4_F32; pop tiles are staged into LDS with the Tensor Data
// Mover (TENSOR_LOAD_TO_LDS) when the builtin is available.
// ---------------------------------------------------------------------------

#define EPS_BN 1e-5f
#define DCH    32               // feature channels D
#define KNN    32               // neighbors k
#define SEQL   256              // L
#define NORM33 (1.0f / 33.0f)   // deg is exactly k+1 = 33 for every node
#define CHUNK  512              // kNN column chunk (LDS-resident distances)
#define QTILE  16               // query rows per kNN workgroup

typedef __attribute__((ext_vector_type(2))) float v2f;
typedef __attribute__((ext_vector_type(8))) float v8f;
typedef __attribute__((ext_vector_type(4))) unsigned int u32x4;
typedef __attribute__((ext_vector_type(8))) int i32x8;
typedef __attribute__((ext_vector_type(4))) int i32x4;

__device__ __forceinline__ float silu(float v) {
    return v / (1.0f + __expf(-v));
}

// ---------------------------------------------------------------- init ------
__global__ void k_init_stats(float* stats) {
    if (threadIdx.x < 4) stats[threadIdx.x] = 0.0f;
}

// --------------------------------------------- global pop mean / meansq -----
__global__ void k_pop_stats(const float* __restrict__ pop, int total,
                            float* __restrict__ stats) {
    __shared__ float s_sum[256];
    __shared__ float s_sq[256];
    float a = 0.0f, b = 0.0f;
    for (int i = blockIdx.x * blockDim.x + threadIdx.x; i < total;
         i += gridDim.x * blockDim.x) {
        float v = pop[i];
        a += v;
        b += v * v;
    }
    s_sum[threadIdx.x] = a;
    s_sq[threadIdx.x]  = b;
    __syncthreads();
    for (int off = 128; off > 0; off >>= 1) {
        if (threadIdx.x < off) {
            s_sum[threadIdx.x] += s_sum[threadIdx.x + off];
            s_sq[threadIdx.x]  += s_sq[threadIdx.x + off];
        }
        __syncthreads();
    }
    if (threadIdx.x == 0) {
        atomicAdd(&stats[0], s_sum[0]);
        atomicAdd(&stats[1], s_sq[0]);
    }
}

// -------------------------------- fused conv+BN+SiLU+mean_L -> x, sq --------
// One wave (32 lanes == 32 channels) per row. Block = 256 threads = 8 rows.
// The 8x256 f32 pop tile is DMA'd into LDS by the Tensor Data Mover.
__global__ void k_features(const float* __restrict__ pop,
                           const float* __restrict__ conv_w,
                           const float* __restrict__ conv_b,
                           const float* __restrict__ bn_gamma,
                           const float* __restrict__ bn_beta,
                           const float* __restrict__ stats,
                           float* __restrict__ x,
                           float* __restrict__ sqn, int n) {
    __shared__ float rowbuf[8][SEQL];
    const int wid  = threadIdx.x >> 5;     // wave within block
    const int lane = threadIdx.x & 31;     // channel d
    const int row0 = blockIdx.x * 8;
    const int row  = row0 + wid;

#if __has_builtin(__builtin_amdgcn_tensor_load_to_lds)
    // ---- TDM tile load: pop[row0 .. row0+7][0..255] -> rowbuf --------------
    if (wid == 0) {
        const unsigned long long ga =
            (unsigned long long)(const void*)(pop + (size_t)row0 * SEQL);
        const unsigned int lds = (unsigned int)(unsigned long long)(void*)&rowbuf[0][0];
        const unsigned int dim0 = SEQL;                 // tensor width (elems)
        const unsigned int dim1 = (unsigned int)(n - row0); // rows left (OOB->0)
        u32x4 g0;
        g0[0] = 1u;                                          // count=1, user D#
        g0[1] = lds;                                         // lds_addr (bytes)
        g0[2] = (unsigned int)(ga & 0xFFFFFFFFu);            // global_addr[31:0]
        g0[3] = (unsigned int)((ga >> 32) & 0x1FFFFFFu)      // global_addr[56:32]
              | (2u << 30);                                  // type = 2 (image)
        i32x8 g1;
        g1[0] = (int)(2u << 16);                             // data_size = 4B
        g1[1] = (int)((dim0 & 0xFFFFu) << 16);               // tensor_dim0 lo16
        g1[2] = (int)(((dim0 >> 16) & 0xFFFFu)               // tensor_dim0 hi16
              |       ((dim1 & 0xFFFFu) << 16));             // tensor_dim1 lo16
        g1[3] = (int)(((dim1 >> 16) & 0xFFFFu)               // tensor_dim1 hi16
              |       ((unsigned)SEQL << 16));               // tile_dim0 = 256
        g1[4] = 8;                                           // tile_dim1 = 8
        g1[5] = SEQL;                                        // dim0_stride lo32
        g1[6] = 0;                                           // strides hi
        g1[7] = 0;
        const i32x4 z4 = {0, 0, 0, 0};                       // 2-D: groups 2/3 off
        const i32x8 z8 = {0, 0, 0, 0, 0, 0, 0, 0};
        __builtin_amdgcn_tensor_load_to_lds(g0, g1, z4, z4, z8, 0);
        __builtin_amdgcn_s_wait_tensorcnt(0);
    }
    __syncthreads();
#else
    if (row < n) {
#pragma unroll
        for (int t = 0; t < SEQL / 32; ++t)
            rowbuf[wid][lane + 32 * t] = pop[row * SEQL + lane + 32 * t];
    }
    __syncthreads();
#endif
    if (row >= n) return;

    const float inv_total = 1.0f / (float)(n * SEQL);
    const float mu   = stats[0] * inv_total;
    const float varp = stats[1] * inv_total - mu * mu;

    const float wd = conv_w[lane];                   // (D,1,1) flat
    const float Ad = bn_gamma[lane] * wd * rsqrtf(wd * wd * varp + EPS_BN);
    const float Bd = bn_beta[lane];
    (void)conv_b;   // conv bias cancels exactly inside BN

    float acc = 0.0f;
#pragma unroll 8
    for (int l = 0; l < SEQL; ++l)
        acc += silu(Ad * (rowbuf[wid][l] - mu) + Bd);
    const float xv = acc * (1.0f / (float)SEQL);
    x[row * DCH + lane] = xv;

    float s = xv * xv;
#pragma unroll
    for (int off = 16; off > 0; off >>= 1)
        s += __shfl_xor(s, off, 32);
    if (lane == 0) sqn[row] = s;
}

// --------------------------- exact kNN (k=32) via WMMA Gram matrix ----------
// One workgroup per 16 query rows. Per 512-column chunk:
//   Phase A: 8 waves compute 16x16 distance tiles with 8x V_WMMA_F32_16X16X4
//            each (for X.X^T the B fragment of column tile j is exactly the
//            A fragment of row tile j), writing d2 into LDS.
//   Phase B: each wave owns 2 query rows; threshold scan + rare serialized
//            insertion into a 32-entry LDS candidate list (wave-private).
__global__ void __launch_bounds__(256)
k_knn_wmma(const float* __restrict__ x,
           const float* __restrict__ sqn,
           int* __restrict__ nbr, int n) {
    __shared__ float distT[QTILE][CHUNK];   // 32 KB
    __shared__ float topd[QTILE][KNN];      // 2 KB
    __shared__ int   topi[QTILE][KNN];      // 2 KB

    const int wid  = threadIdx.x >> 5;
    const int lane = threadIdx.x & 31;
    const int q0   = blockIdx.x * QTILE;

    const int m  = lane & 15;
    const int kh = lane >> 4;

    for (int e = threadIdx.x; e < QTILE * KNN; e += 256) {
        ((float*)topd)[e] = INFINITY;
        ((int*)topi)[e]   = 0;
    }

    // A fragments for the 16 query rows (reused for every column tile)
    v2f afrag[8];
    {
        int qa = q0 + m;
        if (qa >= n) qa = n - 1;
#pragma unroll
        for (int kk = 0; kk < 8; ++kk) {
            const int kb = kk * 4 + kh * 2;
            afrag[kk].x = x[qa * DCH + kb];
            afrag[kk].y = x[qa * DCH + kb + 1];
        }
    }
    // query sq-norms for this lane's 8 C-rows
    float sqq[8];
#pragma unroll
    for (int r = 0; r < 8; ++r) {
        const int qr = q0 + kh * 8 + r;
        sqq[r] = (qr < n) ? sqn[qr] : 0.0f;
    }

    // per-wave selection state: wave owns query rows 2*wid and 2*wid+1
    const int rowA = 2 * wid;
    const int rowB = 2 * wid + 1;
    float worstV[2] = {INFINITY, INFINITY};
    int   slot[2]   = {0, 0};

    const int nChunks = (n + CHUNK - 1) / CHUNK;
    for (int c = 0; c < nChunks; ++c) {
        const int base = c * CHUNK;
        __syncthreads();                     // distT free / top lists ready

        // ---- Phase A: distances via WMMA -------------------------------
        for (int t = wid; t < CHUNK / 16; t += 8) {
            const int j0 = base + t * 16;
            int jb = j0 + m;
            if (jb >= n) jb = n - 1;
            __builtin_prefetch(x + (jb + 16) * DCH, 0, 0);   // global_prefetch_b8
            v8f cacc = {};
#pragma unroll
            for (int kk = 0; kk < 8; ++kk) {
                const int kb = kk * 4 + kh * 2;
                v2f b;
                b.x = x[jb * DCH + kb];
                b.y = x[jb * DCH + kb + 1];
                cacc = __builtin_amdgcn_wmma_f32_16x16x4_f32(
                    false, afrag[kk], false, b, (short)0, cacc, false, false);
            }
            const int j = j0 + m;
            const float sqj = (j < n) ? sqn[j] : 0.0f;
#pragma unroll
            for (int r = 0; r < 8; ++r) {
                const int row = kh * 8 + r;
                float d = sqq[r] + sqj - 2.0f * cacc[r];
                if (j >= n || j == q0 + row) d = INFINITY;
                distT[row][t * 16 + m] = d;
            }
        }
        __syncthreads();

        // ---- Phase B: threshold scan + rare insertion ------------------
        for (int half = 0; half < 2; ++half) {
            const int r  = (half == 0) ? rowA : rowB;
            float worst  = worstV[half];
            int   wslot  = slot[half];
            for (int lc = lane; lc < CHUNK; lc += 32) {
                const float d = distT[r][lc];
                const int   j = base + lc;
                unsigned long long mask = __ballot(d < worst);
                while (mask) {
                    const int src = __builtin_ctzll(mask);
                    mask &= mask - 1;
                    const float dC = __shfl(d, src, 32);
                    const int   jC = __shfl(j, src, 32);
                    if (dC < worst) {          // re-check: worst tightens
                        if (lane == 0) {
                            topd[r][wslot] = dC;
                            topi[r][wslot] = jC;
                        }
                        // DS ops of a wave are in-order; fence the counter
                        // and the compiler before re-reading the list.
                        __asm__ volatile("s_wait_dscnt 0x0" ::: "memory");
                        float v = topd[r][lane];
                        int   s = lane;
#pragma unroll
                        for (int off = 16; off > 0; off >>= 1) {
                            const float ov = __shfl_xor(v, off, 32);
                            const int   os = __shfl_xor(s, off, 32);
                            if (ov > v) { v = ov; s = os; }
                        }
                        worst = v;
                        wslot = s;
                    }
                }
            }
            worstV[half] = worst;
            slot[half]   = wslot;
        }
    }
    __syncthreads();

    if (q0 + rowA < n) nbr[(q0 + rowA) * KNN + lane] = topi[rowA][lane];
    if (q0 + rowB < n) nbr[(q0 + rowB) * KNN + lane] = topi[rowB][lane];
}

// ----------------------------------------------- h = x @ W via WMMA f32 -----
// W is 32x32 row-major. One wave computes a 16-row x 32-col C block.
__global__ void k_matmul_wmma(const float* __restrict__ x,
                              const float* __restrict__ W,
                              float* __restrict__ h, int n) {
    const int wid  = threadIdx.x >> 5;
    const int lane = threadIdx.x & 31;
    const int tile = blockIdx.x * 8 + wid;
    const int row0 = tile * 16;
    if (row0 >= n) return;

    const int m  = lane & 15;
    const int kh = lane >> 4;

    v8f c0 = {};
    v8f c1 = {};
#pragma unroll
    for (int kk = 0; kk < 8; ++kk) {
        const int kb = kk * 4 + kh * 2;
        const int ra = (row0 + m < n) ? (row0 + m) : (n - 1);
        v2f a;
        a.x = x[ra * DCH + kb];
        a.y = x[ra * DCH + kb + 1];
        v2f b0, b1;
        b0.x = W[kb * DCH + m];
        b0.y = W[(kb + 1) * DCH + m];
        b1.x = W[kb * DCH + 16 + m];
        b1.y = W[(kb + 1) * DCH + 16 + m];
        c0 = __builtin_amdgcn_wmma_f32_16x16x4_f32(false, a, false, b0,
                                                   (short)0, c0, false, false);
        c1 = __builtin_amdgcn_wmma_f32_16x16x4_f32(false, a, false, b1,
                                                   (short)0, c1, false, false);
    }

#pragma unroll
    for (int r = 0; r < 8; ++r) {
        const int row = row0 + kh * 8 + r;
        if (row < n) {
            h[row * DCH + m]      = c0[r];
            h[row * DCH + 16 + m] = c1[r];
        }
    }
}

// -------------------------------- zero-pad out_w (32x3) into 32x32 ----------
__global__ void k_pad_w(const float* __restrict__ out_w,
                        float* __restrict__ Wpad) {
    const int idx = blockIdx.x * blockDim.x + threadIdx.x;
    if (idx < DCH * DCH) {
        const int d = idx >> 5, c = idx & 31;
        Wpad[idx] = (c < 3) ? out_w[d * 3 + c] : 0.0f;
    }
}

// ----------------- GCN aggregate: x' = silu( (h[i]+sum_nbr h[j])/33 + b ) ---
__global__ void k_aggregate(const float* __restrict__ h,
                            const int* __restrict__ nbr,
                            const float* __restrict__ bias,
                            float* __restrict__ xo, int n) {
    const int wid  = threadIdx.x >> 5;
    const int lane = threadIdx.x & 31;
    const int i    = blockIdx.x * 8 + wid;
    if (i >= n) return;

    __builtin_prefetch(nbr + i * KNN, 0, 0);
    float val = h[i * DCH + lane];
#pragma unroll 4
    for (int t = 0; t < KNN; ++t) {
        const int j = nbr[i * KNN + t];
        val += h[j * DCH + lane];
    }
    xo[i * DCH + lane] = silu(val * NORM33 + bias[lane]);
}

// ------------- final layer: aggregate 3-channel logits + softmax ------------
__global__ void k_final(const float* __restrict__ h,   // padded 32-ch, 0..2 valid
                        const int* __restrict__ nbr,
                        const float* __restrict__ out_b,
                        float* __restrict__ out, int n) {
    const int i = blockIdx.x * blockDim.x + threadIdx.x;
    if (i >= n) return;

    float a0 = h[i * DCH + 0], a1 = h[i * DCH + 1], a2 = h[i * DCH + 2];
    for (int t = 0; t < KNN; ++t) {
        const int j = nbr[i * KNN + t];
        a0 += h[j * DCH + 0];
        a1 += h[j * DCH + 1];
        a2 += h[j * DCH + 2];
    }
    float z0 = a0 * NORM33 + out_b[0];
    float z1 = a1 * NORM33 + out_b[1];
    float z2 = a2 * NORM33 + out_b[2];
    float zm = fmaxf(z0, fmaxf(z1, z2));
    float e0 = __expf(z0 - zm), e1 = __expf(z1 - zm), e2 = __expf(z2 - zm);
    float inv = 1.0f / (e0 + e1 + e2);
    out[i * 3 + 0] = e0 * inv;
    out[i * 3 + 1] = e1 * inv;
    out[i * 3 + 2] = e2 * inv;
}

// ---------------------------------------------------------------------------
extern "C" void kernel_launch(void* const* d_in, const int* in_sizes, int n_in,
                              void* d_out, int out_size, void* d_ws, size_t ws_size,
                              hipStream_t stream) {
    const float* pop      = (const float*)d_in[0];
    const float* conv_w   = (const float*)d_in[1];
    const float* conv_b   = (const float*)d_in[2];
    const float* bn_gamma = (const float*)d_in[3];
    const float* bn_beta  = (const float*)d_in[4];
    const float* gcn_ws   = (const float*)d_in[5];   // (3,32,32)
    const float* gcn_bs   = (const float*)d_in[6];   // (3,32)
    const float* out_w    = (const float*)d_in[7];   // (32,3)
    const float* out_b    = (const float*)d_in[8];   // (3,)
    float* out = (float*)d_out;

    const int n = in_sizes[0] / SEQL;                // 10000
    (void)n_in; (void)out_size;

    char* ws = (char*)d_ws;
    size_t off = 0;
    auto carve = [&](size_t bytes) -> char* {
        char* p = ws + off;
        off = (off + bytes + 255) & ~(size_t)255;
        return p;
    };
    float* stats = (float*)carve(4 * sizeof(float));
    float* xA    = (float*)carve((size_t)n * DCH * sizeof(float));
    float* xB    = (float*)carve((size_t)n * DCH * sizeof(float));
    float* hbuf  = (float*)carve((size_t)n * DCH * sizeof(float));
    float* sqn   = (float*)carve((size_t)n * sizeof(float));
    int*   nbr   = (int*)  carve((size_t)n * KNN * sizeof(int));
    float* Wpad  = (float*)carve(DCH * DCH * sizeof(float));
    (void)ws_size;

    // 1. global pop statistics
    k_init_stats<<<1, 32, 0, stream>>>(stats);
    {
        const int total = n * SEQL;
        int grid = (total + 255) / 256;
        if (grid > 2048) grid = 2048;
        k_pop_stats<<<grid, 256, 0, stream>>>(pop, total, stats);
    }

    // 2. fused conv/BN/SiLU/mean features + row norms (TDM tile staging)
    k_features<<<(n + 7) / 8, 256, 0, stream>>>(
        pop, conv_w, conv_b, bn_gamma, bn_beta, stats, xA, sqn, n);

    // 3. exact kNN (WMMA Gram matrix + threshold selection)
    k_knn_wmma<<<(n + QTILE - 1) / QTILE, 256, 0, stream>>>(xA, sqn, nbr, n);

    // 4. three GCN layers (WMMA matmul + gather aggregate + SiLU)
    const int mmGrid = (((n + 15) / 16) + 7) / 8;
    float* xcur = xA;
    float* xnxt = xB;
    for (int g = 0; g < 3; ++g) {
        k_matmul_wmma<<<mmGrid, 256, 0, stream>>>(xcur, gcn_ws + g * DCH * DCH,
                                                  hbuf, n);
        k_aggregate<<<(n + 7) / 8, 256, 0, stream>>>(hbuf, nbr,
                                                     gcn_bs + g * DCH, xnxt, n);
        float* t = xcur; xcur = xnxt; xnxt = t;
    }

    // 5. output layer (pad 32x3 -> 32x32, reuse WMMA matmul) + softmax
    k_pad_w<<<4, 256, 0, stream>>>(out_w, Wpad);
    k_matmul_wmma<<<mmGrid, 256, 0, stream>>>(xcur, Wpad, hbuf, n);
    k_final<<<(n + 255) / 256, 256, 0, stream>>>(hbuf, nbr, out_b, out, n);
}